// JobRecGraphSAGE_84533546320019
// MI455X (gfx1250) — compile-verified
//
#include <hip/hip_runtime.h>

typedef __attribute__((ext_vector_type(2))) float v2f;
typedef __attribute__((ext_vector_type(8))) float v8f;

#define N_U 50000
#define N_J 20000
#define N_S 5000
#define HID 128
#define OUT 64
#define E_UJ 800000
#define E_JU 800000
#define E_JS 300000
#define E_SJ 300000

// ---------------------------------------------------------------------------
// Small utility kernels
// ---------------------------------------------------------------------------
__global__ void fill_zero_int(int* __restrict__ p, int n) {
    int i = blockIdx.x * blockDim.x + threadIdx.x;
    int stride = gridDim.x * blockDim.x;
    for (; i < n; i += stride) p[i] = 0;
}

__global__ void count_deg(const int* __restrict__ dst, int E, int* __restrict__ deg) {
    int i = blockIdx.x * blockDim.x + threadIdx.x;
    if (i < E) atomicAdd(&deg[dst[i]], 1);
}

// single-workgroup exclusive scan: deg[0..n) -> rowptr[0..n], cursor[0..n)
__global__ void scan_excl(const int* __restrict__ deg, int* __restrict__ rowptr,
                          int* __restrict__ cursor, int n) {
    __shared__ int lsum[1024];
    const int t = threadIdx.x;
    const int chunk = (n + 1023) / 1024;
    const int lo = t * chunk;
    const int hi = (lo + chunk < n) ? (lo + chunk) : n;
    int s = 0;
    for (int i = lo; i < hi; ++i) s += deg[i];
    lsum[t] = s;
    __syncthreads();
    for (int ofs = 1; ofs < 1024; ofs <<= 1) {   // Hillis-Steele inclusive scan
        int v = lsum[t];
        int add = (t >= ofs) ? lsum[t - ofs] : 0;
        __syncthreads();
        lsum[t] = v + add;
        __syncthreads();
    }
    int base = (t == 0) ? 0 : lsum[t - 1];
    for (int i = lo; i < hi; ++i) {
        rowptr[i] = base;
        cursor[i] = base;
        base += deg[i];
    }
    if (t == 0) rowptr[n] = lsum[1023];
}

__global__ void fill_cols(const int* __restrict__ src, const int* __restrict__ dst,
                          int E, int* __restrict__ cursor, int* __restrict__ col) {
    int i = blockIdx.x * blockDim.x + threadIdx.x;
    if (i < E) {
        int p = atomicAdd(&cursor[dst[i]], 1);
        col[p] = src[i];
    }
}

// ---------------------------------------------------------------------------
// Repack W[d_in][d_out] into B-fragment-native layout so one aligned b64 load
// yields a lane's v2f B fragment:
//   Wt[((q*d_out + c)*2 + h)*2 + j] = W[(4q + 2h + j)*d_out + c]
// (q = k-quad, h = lane half, j = fragment VGPR index)
// ---------------------------------------------------------------------------
__global__ void repack_w(const float* __restrict__ W, float* __restrict__ Wt,
                         int d_in, int d_out) {
    int i = blockIdx.x * blockDim.x + threadIdx.x;
    int total = d_in * d_out;
    if (i >= total) return;
    int k = i / d_out, c = i % d_out;
    int q = k >> 2, rem = k & 3;
    int h = rem >> 1, j = rem & 1;
    Wt[(((size_t)q * d_out + c) * 2 + h) * 2 + j] = W[i];
}

// ---------------------------------------------------------------------------
// Gather-mean: one wave per dst row, vectorized b128/b64 row loads, 2-deep
// neighbor unroll so two gathers are in flight. All tables L2-resident.
// ---------------------------------------------------------------------------
template <int D>
__global__ void gather_mean(const float* __restrict__ xsrc,
                            const int* __restrict__ rowptr,
                            const int* __restrict__ col,
                            float* __restrict__ agg, int N) {
    const int lane = threadIdx.x & 31;
    const int row  = blockIdx.x * (blockDim.x >> 5) + (threadIdx.x >> 5);
    if (row >= N) return;
    const int beg = rowptr[row];
    const int end = rowptr[row + 1];
    const float inv = (end > beg) ? (1.0f / (float)(end - beg)) : 0.0f;

    if constexpr (D == 128) {
        const float4* xb = (const float4*)xsrc;       // 32 float4 per row
        float4 acc = {0.f, 0.f, 0.f, 0.f};
        int e = beg;
        for (; e + 1 < end; e += 2) {
            const int s0 = col[e];
            const int s1 = col[e + 1];
            float4 v0 = xb[(size_t)s0 * 32 + lane];
            float4 v1 = xb[(size_t)s1 * 32 + lane];
            acc.x += v0.x; acc.y += v0.y; acc.z += v0.z; acc.w += v0.w;
            acc.x += v1.x; acc.y += v1.y; acc.z += v1.z; acc.w += v1.w;
        }
        if (e < end) {
            const int s0 = col[e];
            float4 v0 = xb[(size_t)s0 * 32 + lane];
            acc.x += v0.x; acc.y += v0.y; acc.z += v0.z; acc.w += v0.w;
        }
        float4 o = {acc.x * inv, acc.y * inv, acc.z * inv, acc.w * inv};
        ((float4*)agg)[(size_t)row * 32 + lane] = o;
    } else {                                          // D == 64
        const float2* xb = (const float2*)xsrc;       // 32 float2 per row
        float2 acc = {0.f, 0.f};
        int e = beg;
        for (; e + 1 < end; e += 2) {
            const int s0 = col[e];
            const int s1 = col[e + 1];
            float2 v0 = xb[(size_t)s0 * 32 + lane];
            float2 v1 = xb[(size_t)s1 * 32 + lane];
            acc.x += v0.x; acc.y += v0.y;
            acc.x += v1.x; acc.y += v1.y;
        }
        if (e < end) {
            const int s0 = col[e];
            float2 v0 = xb[(size_t)s0 * 32 + lane];
            acc.x += v0.x; acc.y += v0.y;
        }
        float2 o = {acc.x * inv, acc.y * inv};
        ((float2*)agg)[(size_t)row * 32 + lane] = o;
    }
}

// ---------------------------------------------------------------------------
// One software-pipelined GEMM term: c[NT] += x(16 rows) @ W (repacked).
// Fragments for k-step q+1 are fetched before the WMMAs of step q issue, so
// each WMMA's wait covers loads issued a full iteration earlier.
// ---------------------------------------------------------------------------
template <int NT>
__device__ __forceinline__ void gemm_term(v8f* __restrict__ c,
                                          const float* __restrict__ xrow,
                                          const float2* __restrict__ Wt2,
                                          int d_in, int d_out,
                                          int ncol0, int half, float scale) {
    const int nq = d_in >> 2;
    v2f a_cur;
    {
        const int ka = 2 * half;
        a_cur[0] = xrow[ka] * scale;
        a_cur[1] = xrow[ka + 1] * scale;
    }
    float2 b_cur[NT];
#pragma unroll
    for (int t = 0; t < NT; ++t)
        b_cur[t] = Wt2[((size_t)(ncol0 + 16 * t)) * 2 + half];

    for (int q = 0; q < nq; ++q) {
        const int qn = (q + 1 < nq) ? (q + 1) : q;   // harmless re-read on last
        v2f a_nxt;
        const int ka = (qn << 2) + 2 * half;
        a_nxt[0] = xrow[ka] * scale;
        a_nxt[1] = xrow[ka + 1] * scale;
        float2 b_nxt[NT];
#pragma unroll
        for (int t = 0; t < NT; ++t)
            b_nxt[t] = Wt2[((size_t)(qn * d_out + ncol0 + 16 * t)) * 2 + half];

#pragma unroll
        for (int t = 0; t < NT; ++t) {
            v2f b;
            b[0] = b_cur[t].x;
            b[1] = b_cur[t].y;
            c[t] = __builtin_amdgcn_wmma_f32_16x16x4_f32(false, a_cur, false, b,
                                                         (short)0, c[t], false, false);
        }
        a_cur = a_nxt;
#pragma unroll
        for (int t = 0; t < NT; ++t) b_cur[t] = b_nxt[t];
    }
}

// ---------------------------------------------------------------------------
// SAGE linear: out = agg @ Wl + bl + xdst @ Wr   (mean already applied)
// One wave computes a 16x(16*NT) tile: NT independent v8f accumulators
// pipeline through the XDL unit; A fragment shared across column tiles.
// Rows >= N clamped+zero-scaled: no OOB loads, EXEC all-ones across WMMA.
// ---------------------------------------------------------------------------
template <int NT>
__global__ void sage_gemm_wmma(const float* __restrict__ agg,
                               const float* __restrict__ xdst,
                               const float* __restrict__ WlT,
                               const float* __restrict__ bl,
                               const float* __restrict__ WrT,
                               float* __restrict__ out,
                               int N, int d_src, int d_dst, int d_out,
                               int do_acc, int do_relu) {
    const int lane     = threadIdx.x & 31;
    const int wave     = blockIdx.x * (blockDim.x >> 5) + (threadIdx.x >> 5);
    const int n_groups = (d_out >> 4) / NT;          // column-tile groups
    const int tiles_m  = (N + 15) >> 4;
    if (wave >= tiles_m * n_groups) return;
    const int tm   = wave / n_groups;
    const int tg   = wave % n_groups;
    const int half = lane >> 4;
    const int r    = lane & 15;

    const int  arow   = tm * 16 + r;
    const bool rv     = arow < N;
    const int  arow_s = rv ? arow : 0;               // clamp: no OOB loads
    const float scale = rv ? 1.0f : 0.0f;
    const int  ncol0  = tg * (16 * NT) + r;

    v8f c[NT];
#pragma unroll
    for (int t = 0; t < NT; ++t) c[t] = (v8f){0.f,0.f,0.f,0.f,0.f,0.f,0.f,0.f};

    gemm_term<NT>(c, agg  + (size_t)arow_s * d_src, (const float2*)WlT,
                  d_src, d_out, ncol0, half, scale);
    gemm_term<NT>(c, xdst + (size_t)arow_s * d_dst, (const float2*)WrT,
                  d_dst, d_out, ncol0, half, scale);

#pragma unroll
    for (int t = 0; t < NT; ++t) {
        const int ncol = ncol0 + t * 16;
        const float bias = bl[ncol];
#pragma unroll
        for (int v = 0; v < 8; ++v) {
            const int orow = tm * 16 + v + 8 * half;
            if (orow < N) {
                float* op = out + (size_t)orow * d_out + ncol;
                float val = c[t][v] + bias;
                if (do_acc)  val += *op;
                if (do_relu) val = fmaxf(val, 0.0f);
                *op = val;
            }
        }
    }
}

// ---------------------------------------------------------------------------
extern "C" void kernel_launch(void* const* d_in, const int* in_sizes, int n_in,
                              void* d_out, int out_size, void* d_ws, size_t ws_size,
                              hipStream_t stream) {
    const float* x_user  = (const float*)d_in[0];
    const float* x_job   = (const float*)d_in[1];
    const float* x_skill = (const float*)d_in[2];

    const float* l1_uj_Wl = (const float*)d_in[3];
    const float* l1_uj_bl = (const float*)d_in[4];
    const float* l1_uj_Wr = (const float*)d_in[5];
    const float* l2_uj_Wl = (const float*)d_in[6];
    const float* l2_uj_bl = (const float*)d_in[7];
    const float* l2_uj_Wr = (const float*)d_in[8];

    const float* l1_ju_Wl = (const float*)d_in[9];
    const float* l1_ju_bl = (const float*)d_in[10];
    const float* l1_ju_Wr = (const float*)d_in[11];
    const float* l2_ju_Wl = (const float*)d_in[12];
    const float* l2_ju_bl = (const float*)d_in[13];
    const float* l2_ju_Wr = (const float*)d_in[14];

    const float* l1_js_Wl = (const float*)d_in[15];
    const float* l1_js_bl = (const float*)d_in[16];
    const float* l1_js_Wr = (const float*)d_in[17];
    const float* l2_js_Wl = (const float*)d_in[18];
    const float* l2_js_bl = (const float*)d_in[19];
    const float* l2_js_Wr = (const float*)d_in[20];

    const float* l1_sj_Wl = (const float*)d_in[21];
    const float* l1_sj_bl = (const float*)d_in[22];
    const float* l1_sj_Wr = (const float*)d_in[23];
    const float* l2_sj_Wl = (const float*)d_in[24];
    const float* l2_sj_bl = (const float*)d_in[25];
    const float* l2_sj_Wr = (const float*)d_in[26];

    const int* e1_src = (const int*)d_in[27];
    const int* e1_dst = (const int*)d_in[28];
    const int* e2_src = (const int*)d_in[29];
    const int* e2_dst = (const int*)d_in[30];
    const int* e3_src = (const int*)d_in[31];
    const int* e3_dst = (const int*)d_in[32];
    const int* e4_src = (const int*)d_in[33];
    const int* e4_dst = (const int*)d_in[34];

    // --- workspace layout (floats first, then ints) ---
    float* wsf = (float*)d_ws;
    size_t off = 0;
    float* agg_ja = wsf + off; off += (size_t)N_J * 128;
    float* agg_jb = wsf + off; off += (size_t)N_J * 128;
    float* agg_u  = wsf + off; off += (size_t)N_U * 128;
    float* agg_s  = wsf + off; off += (size_t)N_S * 128;
    float* h_u    = wsf + off; off += (size_t)N_U * HID;
    float* h_j    = wsf + off; off += (size_t)N_J * HID;
    float* h_s    = wsf + off; off += (size_t)N_S * HID;

    // repacked weights (fragment-native layout)
    auto walloc = [&](int d_in_, int d_out_) { float* p = wsf + off; off += (size_t)d_in_ * d_out_; return p; };
    float* t1_uj_Wl = walloc(128, HID); float* t1_uj_Wr = walloc(128, HID);
    float* t1_ju_Wl = walloc(128, HID); float* t1_ju_Wr = walloc(128, HID);
    float* t1_js_Wl = walloc(128, HID); float* t1_js_Wr = walloc(64,  HID);
    float* t1_sj_Wl = walloc(64,  HID); float* t1_sj_Wr = walloc(128, HID);
    float* t2_uj_Wl = walloc(128, OUT); float* t2_uj_Wr = walloc(128, OUT);
    float* t2_ju_Wl = walloc(128, OUT); float* t2_ju_Wr = walloc(128, OUT);
    float* t2_js_Wl = walloc(128, OUT); float* t2_js_Wr = walloc(128, OUT);
    float* t2_sj_Wl = walloc(128, OUT); float* t2_sj_Wr = walloc(128, OUT);

    int* wsi = (int*)(wsf + off);
    size_t ioff = 0;
    int* rp_ja = wsi + ioff; ioff += N_J + 1;
    int* rp_jb = wsi + ioff; ioff += N_J + 1;
    int* rp_u  = wsi + ioff; ioff += N_U + 1;
    int* rp_s  = wsi + ioff; ioff += N_S + 1;
    int* cu_ja = wsi + ioff; ioff += N_J;
    int* cu_jb = wsi + ioff; ioff += N_J;
    int* cu_u  = wsi + ioff; ioff += N_U;
    int* cu_s  = wsi + ioff; ioff += N_S;
    int* col_ja = wsi + ioff; ioff += E_UJ;
    int* col_jb = wsi + ioff; ioff += E_SJ;
    int* col_u  = wsi + ioff; ioff += E_JU;
    int* col_s  = wsi + ioff; ioff += E_JS;
    int* deg    = wsi + ioff; ioff += N_U;               // reused per edge type

    float* o_u = (float*)d_out;
    float* o_j = o_u + (size_t)N_U * OUT;
    float* o_s = o_j + (size_t)N_J * OUT;

    // --- weight repack (once per call; 16 tiny matrices) ---
    auto repack = [&](const float* W, float* Wt, int din, int dout) {
        int tot = din * dout;
        repack_w<<<(tot + 255) / 256, 256, 0, stream>>>(W, Wt, din, dout);
    };
    repack(l1_uj_Wl, t1_uj_Wl, 128, HID); repack(l1_uj_Wr, t1_uj_Wr, 128, HID);
    repack(l1_ju_Wl, t1_ju_Wl, 128, HID); repack(l1_ju_Wr, t1_ju_Wr, 128, HID);
    repack(l1_js_Wl, t1_js_Wl, 128, HID); repack(l1_js_Wr, t1_js_Wr, 64,  HID);
    repack(l1_sj_Wl, t1_sj_Wl, 64,  HID); repack(l1_sj_Wr, t1_sj_Wr, 128, HID);
    repack(l2_uj_Wl, t2_uj_Wl, 128, OUT); repack(l2_uj_Wr, t2_uj_Wr, 128, OUT);
    repack(l2_ju_Wl, t2_ju_Wl, 128, OUT); repack(l2_ju_Wr, t2_ju_Wr, 128, OUT);
    repack(l2_js_Wl, t2_js_Wl, 128, OUT); repack(l2_js_Wr, t2_js_Wr, 128, OUT);
    repack(l2_sj_Wl, t2_sj_Wl, 128, OUT); repack(l2_sj_Wr, t2_sj_Wr, 128, OUT);

    // --- CSR build (edges identical for both layers: build once per call) ---
    auto build_csr = [&](const int* src, const int* dst, int E, int N,
                         int* rp, int* cur, int* col) {
        fill_zero_int<<<(N + 255) / 256, 256, 0, stream>>>(deg, N);
        count_deg<<<(E + 255) / 256, 256, 0, stream>>>(dst, E, deg);
        scan_excl<<<1, 1024, 0, stream>>>(deg, rp, cur, N);
        fill_cols<<<(E + 255) / 256, 256, 0, stream>>>(src, dst, E, cur, col);
    };
    build_csr(e1_src, e1_dst, E_UJ, N_J, rp_ja, cu_ja, col_ja);
    build_csr(e4_src, e4_dst, E_SJ, N_J, rp_jb, cu_jb, col_jb);
    build_csr(e2_src, e2_dst, E_JU, N_U, rp_u,  cu_u,  col_u);
    build_csr(e3_src, e3_dst, E_JS, N_S, rp_s,  cu_s,  col_s);

    auto gemm = [&](const float* ag, const float* xd,
                    const float* WlT, const float* bl, const float* WrT,
                    float* out, int N, int ds, int dd, int dout,
                    int acc, int relu) {
        int waves  = ((N + 15) / 16) * ((dout / 16) / 4);  // NT=4 col tiles/wave
        int blocks = (waves + 7) / 8;                      // 8 waves/block
        sage_gemm_wmma<4><<<blocks, 256, 0, stream>>>(ag, xd, WlT, bl, WrT, out,
                                                      N, ds, dd, dout, acc, relu);
    };
    auto gblk = [](int N) { return (N + 7) / 8; };         // 8 waves/block

    // ---------------- Layer 1 ----------------
    gather_mean<128><<<gblk(N_J), 256, 0, stream>>>(x_user,  rp_ja, col_ja, agg_ja, N_J);
    gather_mean< 64><<<gblk(N_J), 256, 0, stream>>>(x_skill, rp_jb, col_jb, agg_jb, N_J);
    gather_mean<128><<<gblk(N_U), 256, 0, stream>>>(x_job,   rp_u,  col_u,  agg_u,  N_U);
    gather_mean<128><<<gblk(N_S), 256, 0, stream>>>(x_job,   rp_s,  col_s,  agg_s,  N_S);

    gemm(agg_ja, x_job,   t1_uj_Wl, l1_uj_bl, t1_uj_Wr, h_j, N_J, 128, 128, HID, 0, 0);
    gemm(agg_jb, x_job,   t1_sj_Wl, l1_sj_bl, t1_sj_Wr, h_j, N_J, 64,  128, HID, 1, 1);
    gemm(agg_u,  x_user,  t1_ju_Wl, l1_ju_bl, t1_ju_Wr, h_u, N_U, 128, 128, HID, 0, 1);
    gemm(agg_s,  x_skill, t1_js_Wl, l1_js_bl, t1_js_Wr, h_s, N_S, 128, 64,  HID, 0, 1);

    // ---------------- Layer 2 ----------------
    gather_mean<128><<<gblk(N_J), 256, 0, stream>>>(h_u, rp_ja, col_ja, agg_ja, N_J);
    gather_mean<128><<<gblk(N_J), 256, 0, stream>>>(h_s, rp_jb, col_jb, agg_jb, N_J);
    gather_mean<128><<<gblk(N_U), 256, 0, stream>>>(h_j, rp_u,  col_u,  agg_u,  N_U);
    gather_mean<128><<<gblk(N_S), 256, 0, stream>>>(h_j, rp_s,  col_s,  agg_s,  N_S);

    gemm(agg_ja, h_j, t2_uj_Wl, l2_uj_bl, t2_uj_Wr, o_j, N_J, 128, 128, OUT, 0, 0);
    gemm(agg_jb, h_j, t2_sj_Wl, l2_sj_bl, t2_sj_Wr, o_j, N_J, 128, 128, OUT, 1, 0);
    gemm(agg_u,  h_u, t2_ju_Wl, l2_ju_bl, t2_ju_Wr, o_u, N_U, 128, 128, OUT, 0, 0);
    gemm(agg_s,  h_s, t2_js_Wl, l2_js_bl, t2_js_Wr, o_s, N_S, 128, 128, OUT, 0, 0);
}